// QValueNetMultiHeadAttention_35699768164380
// MI455X (gfx1250) — compile-verified
//
#include <hip/hip_runtime.h>

// ---------------------------------------------------------------------------
// QValueNetMultiHeadAttention — collapsed form for MI455X (gfx1250).
//
// Math: softmax over a SINGLETON axis == 1.0 exactly, so attention weights
// are all-ones and q/k/Wq/Wk/Ws are dead.  The reference reduces to:
//     row[b,:]   = (sum_k values[b,k,:]) @ Wv^T @ Wo^T      ([4,512])
//     out[b,q,:] = row[b,:]  for all q                       ([4,256,512])
//
// Pipeline (all on `stream`, ws scratch, ~6 MB total HBM traffic):
//   K1: vs[b,c] = sum_k values[b,k,c]             (2 MB read, coalesced)
//   K2: tmp     = vs  @ Wv^T   via V_WMMA_F32_16X16X4_F32 (fp32 WMMA)
//   K3: row     = tmp @ Wo^T   via V_WMMA_F32_16X16X4_F32
//   K4: broadcast row over 256 query positions    (2 MB write, b128 stores)
// ---------------------------------------------------------------------------

typedef __attribute__((ext_vector_type(2))) float v2f;
typedef __attribute__((ext_vector_type(8))) float v8f;

#define BATCH 4
#define SEQ_LQ 256
#define SEQ_LK 256
#define NHID 512   // NH
#define VDIM 512   // VS

// --- K1: column-sum of values over the key axis ----------------------------
// grid = BATCH * (VDIM/256) = 8 blocks of 256 threads; thread owns one column.
__global__ void k_reduce_values(const float* __restrict__ values,
                                float* __restrict__ vs) {
    const int b = blockIdx.x >> 1;
    const int c = ((blockIdx.x & 1) << 8) + threadIdx.x;     // 0..511
    const float* p = values + (size_t)b * SEQ_LK * VDIM + c;
    float acc = 0.0f;
#pragma unroll 8
    for (int k = 0; k < SEQ_LK; ++k)
        acc += p[(size_t)k * VDIM];                          // coalesced
    vs[b * VDIM + c] = acc;
}

// --- K2/K3: out[4,N] = A[4,K] @ W[N,K]^T using V_WMMA_F32_16X16X4_F32 ------
// One wave (32 threads) per 16-column tile of N; K consumed 4 at a time.
// A-fragment layout (ISA 7.12.2, 32-bit A 16x4): lanes 0-15 hold M=lane with
// VGPR0=K0,VGPR1=K1; lanes 16-31 hold M=lane-16 with VGPR0=K2,VGPR1=K3.
// B-fragment is the dual (4x16): lanes 0-15 hold N=lane rows K0/K1, lanes
// 16-31 rows K2/K3.  D: VGPR j -> M=j (lanes 0-15), M=j+8 (lanes 16-31).
//
// Rows 4..15 of A are zero-padded BRANCHLESSLY: lanes with l16>=4 re-read
// row 0 (L0-resident, uniform) and multiply the fragment by mask=0.0f.
// This keeps EXEC untouched across the whole K loop (WMMA needs EXEC=all-1).
__global__ void k_gemm4_wmma(const float* __restrict__ A,   // [4, K] row-major
                             const float* __restrict__ W,   // [N, K] row-major
                             float* __restrict__ out,       // [4, N]
                             int K, int N) {
    const int ntile = blockIdx.x;          // 16 output columns per wave
    const int lane  = threadIdx.x;         // 0..31 (wave32)
    const int half  = lane >> 4;           // 0: K+0/K+1 ; 1: K+2/K+3
    const int l16   = lane & 15;
    const int col   = ntile * 16 + l16;    // output column owned by this lane

    const int   arow = (l16 < BATCH) ? l16 : 0;     // one v_cndmask, hoisted
    const float mask = (l16 < BATCH) ? 1.0f : 0.0f; // one v_cndmask, hoisted

    // v2f-indexed base pointers; ka = k0 + 2*half is always even.
    const v2f* __restrict__ ap = (const v2f*)(A + (size_t)arow * K);
    const v2f* __restrict__ wp = (const v2f*)(W + (size_t)col * K);
    const int  hoff = half;                // v2f-granular half offset

    v8f acc = {};                          // C/D accumulator, starts at 0
#pragma unroll 4
    for (int k0 = 0; k0 < K; k0 += 4) {
        const int kv = (k0 >> 1) + hoff;   // v2f index of {ka, ka+1}

        v2f afrag = ap[kv];                // global_load_b64
        afrag.x *= mask;                   // branchless zero-pad rows 4..15
        afrag.y *= mask;

        const v2f bfrag = wp[kv];          // global_load_b64

        // D = A(16x4) * B(4x16) + C  -> v_wmma_f32_16x16x4_f32
        acc = __builtin_amdgcn_wmma_f32_16x16x4_f32(
            /*neg_a=*/false, afrag,
            /*neg_b=*/false, bfrag,
            /*c_mod=*/(short)0, acc,
            /*reuse_a=*/false, /*reuse_b=*/false);
    }

    // Rows M=0..3 live in acc[0..3] on lanes 0-15 (divergence only AFTER all
    // WMMAs have issued).
    if (half == 0) {
        out[0 * N + col] = acc[0];
        out[1 * N + col] = acc[1];
        out[2 * N + col] = acc[2];
        out[3 * N + col] = acc[3];
    }
}

// --- K4: broadcast row[b,:] across the 256 query positions -----------------
// float4 granularity -> global_store_b128, fully coalesced 2 MB write.
__global__ void k_broadcast(const float4* __restrict__ row4,
                            float4* __restrict__ out4) {
    const size_t idx = (size_t)blockIdx.x * blockDim.x + threadIdx.x;
    const int v4 = (int)(idx & (VDIM / 4 - 1));   // 128 float4 per out row
    const int b  = (int)(idx >> 15);              // LQ*VDIM/4 = 32768 = 2^15
    out4[idx] = row4[(b << 7) + v4];              // row slice is L2-resident
}

extern "C" void kernel_launch(void* const* d_in, const int* in_sizes, int n_in,
                              void* d_out, int out_size, void* d_ws, size_t ws_size,
                              hipStream_t stream) {
    (void)in_sizes; (void)n_in; (void)out_size; (void)ws_size;

    // setup_inputs order: queries, keys, values, Wq, Wk, Wv, Ws, Wo
    const float* values = (const float*)d_in[2];
    const float* Wv     = (const float*)d_in[5];
    const float* Wo     = (const float*)d_in[7];

    // scratch layout in d_ws: vs[4*512] | tmp[4*512] | row[4*512]  (24 KB)
    float* vs  = (float*)d_ws;
    float* tmp = vs  + BATCH * NHID;
    float* row = tmp + BATCH * NHID;

    // K1: vs[b,c] = sum_k values[b,k,c]
    k_reduce_values<<<BATCH * (VDIM / 256), 256, 0, stream>>>(values, vs);

    // K2: tmp = vs @ Wv^T   (M=4, K=512, N=512)
    k_gemm4_wmma<<<NHID / 16, 32, 0, stream>>>(vs, Wv, tmp, VDIM, NHID);

    // K3: row = tmp @ Wo^T  (M=4, K=512, N=512)
    k_gemm4_wmma<<<VDIM / 16, 32, 0, stream>>>(tmp, Wo, row, NHID, VDIM);

    // K4: out[b,q,:] = row[b,:]
    k_broadcast<<<(BATCH * SEQ_LQ * VDIM / 4) / 256, 256, 0, stream>>>(
        (const float4*)row, (float4*)d_out);
}